// DLPGNN_31284541784802
// MI455X (gfx1250) — compile-verified
//
#include <hip/hip_runtime.h>
#include <hip/hip_bf16.h>

#define MDIM 4096
#define NDIM 4096

typedef __attribute__((ext_vector_type(2))) float v2f;
typedef __attribute__((ext_vector_type(8))) float v8f;

__device__ __forceinline__ float sigm(float z) { return 1.0f / (1.0f + __expf(-z)); }

// ---------------------------------------------------------------------------
// Row reduction: out[i] = (op_j A[i,j]*v[j]) * (mul ? mul[i] : 1)
// One wave32 per row, float4 streaming loads, shfl tree reduce.
// The shared operand v (16KB) is staged ONCE per block into LDS via
// GLOBAL_LOAD_ASYNC_TO_LDS_B128 (ASYNCcnt path) and served to all 8 waves
// from the DS pipe, keeping the vmem pipe exclusively for the A stream.
// grid: 512 blocks x 256 threads (8 waves/block) -> 4096 rows.
// ---------------------------------------------------------------------------
template <bool IS_MAX>
__global__ void k_rowred(const float* __restrict__ A, const float* __restrict__ v,
                         const float* __restrict__ mul, float* __restrict__ out) {
  __shared__ __align__(16) float vlds[NDIM];

  const int wave = threadIdx.x >> 5;
  const int lane = threadIdx.x & 31;
  const int row  = blockIdx.x * 8 + wave;

  if (v) {
    // Each thread owns 16B per 4KB chunk; 4 chunks cover 16KB.
    // Flat LDS pointer low 32 bits == LDS byte offset (aperture truncation).
    unsigned lds_addr = (unsigned)(size_t)(&vlds[threadIdx.x * 4]);
    unsigned long long ga = (unsigned long long)(v + threadIdx.x * 4);
    asm volatile(
        "global_load_async_to_lds_b128 %0, %1, off offset:0\n\t"
        "global_load_async_to_lds_b128 %0, %1, off offset:4096\n\t"
        "global_load_async_to_lds_b128 %0, %1, off offset:8192\n\t"
        "global_load_async_to_lds_b128 %0, %1, off offset:12288\n\t"
        "s_wait_asynccnt 0x0"
        :: "v"(lds_addr), "v"(ga) : "memory");
    __syncthreads();
  }

  const float4* Ar = reinterpret_cast<const float4*>(A + (size_t)row * NDIM);
  const float4* V4 = reinterpret_cast<const float4*>(vlds);

  float4 acc = IS_MAX ? make_float4(-__builtin_inff(), -__builtin_inff(),
                                    -__builtin_inff(), -__builtin_inff())
                      : make_float4(0.f, 0.f, 0.f, 0.f);
  #pragma unroll 4
  for (int t = 0; t < NDIM / 4 / 32; ++t) {   // 32 iterations
    const int idx = t * 32 + lane;
    float4 a = Ar[idx];                        // global stream (vmem pipe)
    float4 b = v ? V4[idx] : make_float4(1.f, 1.f, 1.f, 1.f);  // ds_load_b128
    if (IS_MAX) {
      acc.x = fmaxf(acc.x, a.x * b.x); acc.y = fmaxf(acc.y, a.y * b.y);
      acc.z = fmaxf(acc.z, a.z * b.z); acc.w = fmaxf(acc.w, a.w * b.w);
    } else {
      acc.x = fmaf(a.x, b.x, acc.x); acc.y = fmaf(a.y, b.y, acc.y);
      acc.z = fmaf(a.z, b.z, acc.z); acc.w = fmaf(a.w, b.w, acc.w);
    }
  }
  float r = IS_MAX ? fmaxf(fmaxf(acc.x, acc.y), fmaxf(acc.z, acc.w))
                   : (acc.x + acc.y) + (acc.z + acc.w);
  #pragma unroll
  for (int off = 16; off > 0; off >>= 1) {
    float o = __shfl_down(r, off, 32);
    r = IS_MAX ? fmaxf(r, o) : r + o;
  }
  if (lane == 0) out[row] = r * (mul ? mul[row] : 1.0f);
}

// ---------------------------------------------------------------------------
// Column reduction partials: part[rb*N + j] = op_{i in rowblock rb} A[i,j]*w[i]
// grid: (4 col-blocks, 32 row-blocks) x 256 threads; thread owns 4 columns.
// Fully coalesced: a block reads 4KB-contiguous row segments.
// ---------------------------------------------------------------------------
template <bool IS_MAX>
__global__ void k_colred(const float* __restrict__ A, const float* __restrict__ w,
                         float* __restrict__ part) {
  const int j0 = (blockIdx.x * blockDim.x + threadIdx.x) * 4;
  const int i0 = blockIdx.y * (MDIM / 32);
  const float* p = A + (size_t)i0 * NDIM + j0;
  float4 acc = IS_MAX ? make_float4(-__builtin_inff(), -__builtin_inff(),
                                    -__builtin_inff(), -__builtin_inff())
                      : make_float4(0.f, 0.f, 0.f, 0.f);
  #pragma unroll 4
  for (int i = 0; i < MDIM / 32; ++i) {       // 128 rows
    float4 a = *reinterpret_cast<const float4*>(p);
    float wi = w ? w[i0 + i] : 1.0f;          // uniform broadcast, L0 hit
    if (IS_MAX) {
      acc.x = fmaxf(acc.x, a.x * wi); acc.y = fmaxf(acc.y, a.y * wi);
      acc.z = fmaxf(acc.z, a.z * wi); acc.w = fmaxf(acc.w, a.w * wi);
    } else {
      acc.x = fmaf(a.x, wi, acc.x); acc.y = fmaf(a.y, wi, acc.y);
      acc.z = fmaf(a.z, wi, acc.z); acc.w = fmaf(a.w, wi, acc.w);
    }
    p += NDIM;
  }
  *reinterpret_cast<float4*>(part + (size_t)blockIdx.y * NDIM + j0) = acc;
}

// ---------------------------------------------------------------------------
// Dual weighted column sum via V_WMMA_F32_16X16X4_F32:
//   p0[rb,j] = sum_{i in rb} A[i,j]*w0[i] ; p1[rb,j] = sum A[i,j]*w1[i]
// D(16x16) = Aop(16x4 = weight tile, rows 0/1 live) x Bop(4x16 = A chunk).
// Bop lane layout: lane(m,h) holds A[r+2h][j0+m], A[r+2h+1][j0+m]
//   -> each load is two contiguous 64B half-wave segments (coalesced).
// Aop lane layout: lane(m,h) holds W[m][2h], W[m][2h+1]; W row0=w0,row1=w1.
// grid: (32, 8) x 256 threads (8 waves/block); wave owns 16 cols x 512 rows.
// ---------------------------------------------------------------------------
__global__ void k_wmma_colsum2(const float* __restrict__ A,
                               const float* __restrict__ w0,
                               const float* __restrict__ w1,
                               float* __restrict__ p0, float* __restrict__ p1) {
  const int wave = threadIdx.x >> 5;
  const int lane = threadIdx.x & 31;
  const int h = lane >> 4;           // K-half selector
  const int m = lane & 15;           // W row (A-op) / column (B-op, D)
  const int j0 = (blockIdx.x * 8 + wave) * 16;
  const int i0 = blockIdx.y * (MDIM / 8);       // 512 rows per row-block

  const float* wp   = (m == 0) ? w0 : w1;       // m>=2 reads w1 but gets zeroed
  const bool   wact = (m < 2);
  const float* bp0 = A + (size_t)(i0 + 2 * h) * NDIM + (j0 + m);
  const float* bp1 = bp0 + NDIM;
  int r0 = i0 + 2 * h;

  v8f d = {};
  for (int s = 0; s < (MDIM / 8) / 4; ++s) {    // 128 WMMA steps, K advances by 4 rows
    v2f bop; bop.x = *bp0; bop.y = *bp1;
    float wv0 = wp[r0], wv1 = wp[r0 + 1];
    v2f aop;
    aop.x = wact ? wv0 : 0.0f;                  // v_cndmask, no divergence
    aop.y = wact ? wv1 : 0.0f;
    d = __builtin_amdgcn_wmma_f32_16x16x4_f32(false, aop, false, bop,
                                              (short)0, d, false, false);
    bp0 += 4 * NDIM; bp1 += 4 * NDIM; r0 += 4;
  }
  if (h == 0) {                                  // D row0 -> d[0], row1 -> d[1], lanes 0..15
    p0[(size_t)blockIdx.y * NDIM + j0 + m] = d[0];
    p1[(size_t)blockIdx.y * NDIM + j0 + m] = d[1];
  }
}

// ------------------------- small elementwise kernels ------------------------
__global__ void k_colsum_combine(const float* __restrict__ part, float* __restrict__ out) {
  int j = blockIdx.x * blockDim.x + threadIdx.x;
  float s = 0.f;
  for (int rb = 0; rb < 32; ++rb) s += part[(size_t)rb * NDIM + j];
  out[j] = s;
}

__global__ void k_right_mask(const float* __restrict__ rowsum, const float* __restrict__ tqr,
                             float* __restrict__ rm, float* __restrict__ one_r,
                             float* __restrict__ y) {
  int i = blockIdx.x * blockDim.x + threadIdx.x;
  float v = rowsum[i], s = 0.f;
  #pragma unroll
  for (int q = 0; q < 4; ++q) s += sigm(fmaf(tqr[2 * q], v, tqr[2 * q + 1]));
  s *= 0.25f;
  rm[i] = s; one_r[i] = 1.0f - s; y[i] = 0.0f;
}

__global__ void k_left_mask_init(const float* __restrict__ part, const float* __restrict__ colsum,
                                 const float* __restrict__ tql, const float* __restrict__ fptr,
                                 float* __restrict__ lm, float* __restrict__ one_l,
                                 float* __restrict__ r, float* __restrict__ b,
                                 float* __restrict__ x) {
  int j = blockIdx.x * blockDim.x + threadIdx.x;
  float mx = -__builtin_inff();
  for (int rb = 0; rb < 32; ++rb) mx = fmaxf(mx, part[(size_t)rb * NDIM + j]);
  float v = colsum[j], s = 0.f;
  #pragma unroll
  for (int q = 0; q < 4; ++q) s += sigm(fmaf(tql[2 * q], v, tql[2 * q + 1]));
  s *= 0.25f;
  float L = fmaxf(s, mx);
  lm[j] = L; one_l[j] = 1.0f - L;
  r[j] = *fptr; b[j] = 1.0f; x[j] = 0.0f;
}

__global__ void k_colmax_combine_mul(const float* __restrict__ part,
                                     const float* __restrict__ mul, float* __restrict__ out) {
  int j = blockIdx.x * blockDim.x + threadIdx.x;
  float mx = -__builtin_inff();
  for (int rb = 0; rb < 32; ++rb) mx = fmaxf(mx, part[(size_t)rb * NDIM + j]);
  out[j] = mx * mul[j];
}

__global__ void k_ydelta(const float* __restrict__ ro, const float* __restrict__ ro_max,
                         const float* __restrict__ one_r, const float* __restrict__ thg_l,
                         float* __restrict__ yd, float* __restrict__ ydro,
                         float* __restrict__ y) {
  int i = blockIdx.x * blockDim.x + threadIdx.x;
  float z = ro[i] - 0.5f * ro_max[i];
  float g = 0.f;
  #pragma unroll
  for (int k = 0; k < 16; ++k) {
    float a = thg_l[3 * k], bb = thg_l[3 * k + 1], c = thg_l[3 * k + 2];
    g = fmaf(a, sigm(fmaf(bb, z, c)), g);
  }
  float orr = one_r[i];
  float ydv = g * orr;
  yd[i] = ydv;
  y[i] = (y[i] + ydv) * orr;
  ydro[i] = ydv / ro[i];
}

__global__ void k_update(const float* __restrict__ p0, const float* __restrict__ p1,
                         const float* __restrict__ one_l, const float* __restrict__ thb_l,
                         const float* __restrict__ fptr, float* __restrict__ r,
                         float* __restrict__ x, float* __restrict__ b) {
  int j = blockIdx.x * blockDim.x + threadIdx.x;
  float t1 = 0.f, t2 = 0.f;
  #pragma unroll
  for (int rb = 0; rb < 8; ++rb) {
    t1 += p0[(size_t)rb * NDIM + j];
    t2 += p1[(size_t)rb * NDIM + j];
  }
  float ol = one_l[j];
  float rv = (r[j] - fmaxf(t1, 0.0f)) * ol;
  r[j] = rv;
  x[j] = (x[j] + b[j] * t2) * ol;
  float f = *fptr, bn = 0.f;
  #pragma unroll
  for (int k = 0; k < 16; ++k) {
    float a = thb_l[4 * k], bb = thb_l[4 * k + 1], c = thb_l[4 * k + 2], dd = thb_l[4 * k + 3];
    float s = sigm(fmaf(bb, rv, c));
    float p = exp2f(-fmaxf(dd * (rv - f), 0.0f));   // ALPHA = 2
    bn = fmaf(a, s * p, bn);
  }
  b[j] = bn * ol;
}

__global__ void k_vfin(const float* __restrict__ h, const float* __restrict__ lm,
                       const float* __restrict__ colsum, float* __restrict__ vfin) {
  int j = blockIdx.x * blockDim.x + threadIdx.x;
  vfin[j] = h[j] * lm[j] / colsum[j];
}

__global__ void k_out(const float* __restrict__ x, const float* __restrict__ y,
                      const float* __restrict__ rm, const float* __restrict__ dualmv,
                      float* __restrict__ out) {
  int i = blockIdx.x * blockDim.x + threadIdx.x;
  out[i] = x[i];                                // pred_primal
  out[NDIM + i] = y[i] + rm[i] + dualmv[i];     // pred_dual
}

// ---------------------------------------------------------------------------
extern "C" void kernel_launch(void* const* d_in, const int* in_sizes, int n_in,
                              void* d_out, int out_size, void* d_ws, size_t ws_size,
                              hipStream_t stream) {
  (void)in_sizes; (void)n_in; (void)out_size; (void)ws_size;
  const float* A   = (const float*)d_in[0];
  const float* thg = (const float*)d_in[1];   // (5,16,3)
  const float* thb = (const float*)d_in[2];   // (5,16,4)
  const float* fp  = (const float*)d_in[3];   // scalar
  const float* h   = (const float*)d_in[4];   // (N,)
  const float* tql = (const float*)d_in[5];   // (4,2)
  const float* tqr = (const float*)d_in[6];   // (4,2)
  float* out = (float*)d_out;

  float* W = (float*)d_ws;                    // all offsets in floats
  float* colsum = W + 0  * 4096;
  float* rtmp   = W + 1  * 4096;   // rowsum, later final dual matvec
  float* rm     = W + 2  * 4096;
  float* one_r  = W + 3  * 4096;
  float* lm     = W + 4  * 4096;
  float* one_l  = W + 5  * 4096;
  float* rvec   = W + 6  * 4096;
  float* bvec   = W + 7  * 4096;
  float* xvec   = W + 8  * 4096;
  float* yvec   = W + 9  * 4096;
  float* ro     = W + 10 * 4096;
  float* ro_t   = W + 11 * 4096;
  float* ro_max = W + 12 * 4096;
  float* yd     = W + 13 * 4096;
  float* ydro   = W + 14 * 4096;
  float* vfin   = W + 15 * 4096;
  float* partA  = W + 16 * 4096;   // 32 x 4096 column-reduction partials
  float* pW0    = W + 48 * 4096;   // 8 x 4096 WMMA partials
  float* pW1    = W + 56 * 4096;   // 8 x 4096

  const dim3 bs(256);
  const dim3 gRow(512);            // row reductions: 8 waves/block, 1 row/wave
  const dim3 gCol(4, 32);          // column reductions: 4 col-blocks x 32 row-blocks
  const dim3 gVec(16);             // 4096-element elementwise
  const dim3 gWm(32, 8);           // WMMA: 256 col-slots x 8 row-blocks

  // Prologue: colsum = A.T @ 1, rowsum = A @ 1
  k_colred<false><<<gCol, bs, 0, stream>>>(A, nullptr, partA);
  k_colsum_combine<<<gVec, bs, 0, stream>>>(partA, colsum);
  k_rowred<false><<<gRow, bs, 0, stream>>>(A, nullptr, nullptr, rtmp);
  k_right_mask<<<gVec, bs, 0, stream>>>(rtmp, tqr, rm, one_r, yvec);
  // case2 = colmax(A[i,j]*rm[i]); left_mask = max(qmix(colsum), case2); init r,b,x
  k_colred<true><<<gCol, bs, 0, stream>>>(A, rm, partA);
  k_left_mask_init<<<gVec, bs, 0, stream>>>(partA, colsum, tql, fp, lm, one_l,
                                            rvec, bvec, xvec);

  for (int l = 0; l < 5; ++l) {
    // ro = (A @ b) * one_r
    k_rowred<false><<<gRow, bs, 0, stream>>>(A, bvec, one_r, ro);
    // ro_t = colmax(A[i,j]*ro[i]) * one_l
    k_colred<true><<<gCol, bs, 0, stream>>>(A, ro, partA);
    k_colmax_combine_mul<<<gVec, bs, 0, stream>>>(partA, one_l, ro_t);
    // ro_max = rowmax(A[i,j]*ro_t[j]) * one_r
    k_rowred<true><<<gRow, bs, 0, stream>>>(A, ro_t, one_r, ro_max);
    // y_delta, y update, y_delta/ro
    k_ydelta<<<gVec, bs, 0, stream>>>(ro, ro_max, one_r, thg + l * 48, yd, ydro, yvec);
    // dual column sums via WMMA: A.T@yd and A.T@(yd/ro)
    k_wmma_colsum2<<<gWm, bs, 0, stream>>>(A, yd, ydro, pW0, pW1);
    k_update<<<gVec, bs, 0, stream>>>(pW0, pW1, one_l, thb + l * 64, fp,
                                      rvec, xvec, bvec);
  }

  // Epilogue: pred_dual = y + rm + A @ (h*lm/colsum)
  k_vfin<<<gVec, bs, 0, stream>>>(h, lm, colsum, vfin);
  k_rowred<false><<<gRow, bs, 0, stream>>>(A, vfin, nullptr, rtmp);
  k_out<<<gVec, bs, 0, stream>>>(xvec, yvec, rm, rtmp, out);
}